// STLinear_76527727280520
// MI455X (gfx1250) — compile-verified
//
#include <hip/hip_runtime.h>

typedef __attribute__((ext_vector_type(2))) float v2f;
typedef __attribute__((ext_vector_type(4))) float v4f;
typedef __attribute__((ext_vector_type(8))) float v8f;

#define B_    32
#define L_    288
#define D_    1024
#define PAST_ 12
#define FUT_  12
#define T_    (L_ - PAST_)   // 276

// ---------------------------------------------------------------------------
// Kernel 1: Wm = Ws * adj   (written directly into its d_out slot; reused by GEMM)
// ---------------------------------------------------------------------------
__global__ __launch_bounds__(256) void wm_mask_kernel(const float* __restrict__ Ws,
                                                      const float* __restrict__ adj,
                                                      float* __restrict__ wm) {
    int i = blockIdx.x * 256 + threadIdx.x;
    wm[i] = Ws[i] * adj[i];
}

// ---------------------------------------------------------------------------
// Kernel 2: copy Wt into its d_out slot (tuple output #2)
// ---------------------------------------------------------------------------
__global__ void wt_copy_kernel(const float* __restrict__ Wt, float* __restrict__ wtOut) {
    int i = threadIdx.x;
    if (i < FUT_ * PAST_) wtOut[i] = Wt[i];
}

// ---------------------------------------------------------------------------
// Kernel 3: out_s[m,e] = sum_d inp[m,d] * Wm[e,d] + bs[e]
//   M = B*L = 9216, N = K = D = 1024. Full-fp32 V_WMMA_F32_16X16X4_F32.
//   Block = 256 threads = 8 waves; block owns a 16-row M tile; wave w owns
//   128 columns as 8 accumulator tiles (A fragment reused 8x per K step).
//
//   K-permutation trick: the reduction over K is order-independent, so apply
//   the SAME physical-k permutation to A and B fragments. Per chunk of 8
//   physical k, lane-half h loads one float4 at (chunk + 4h):
//     WMMA #0 uses .xy  -> physical k {c, c+1, c+4, c+5}
//     WMMA #1 uses .zw  -> physical k {c+2, c+3, c+6, c+7}
//   Hardware pairs A[kk=2h+j] with B[kk=2h+j]; both map to physical c+4h(+2)+j,
//   so every physical k is consumed consistently exactly once per chunk.
//   => 9 b128 loads per 16 WMMAs (vs 18 b64 loads), half the VMEM issues.
//
//   fp32 16x16 C layout: row = vgpr + 8*(lane>>4), col = lane&15.
// ---------------------------------------------------------------------------
__global__ __launch_bounds__(256) void spatial_gemm_kernel(
    const float* __restrict__ inp,   // [9216, 1024]
    const float* __restrict__ wm,    // [1024, 1024]
    const float* __restrict__ bs,    // [1024]
    float* __restrict__ outS)        // [9216, 1024]
{
    const int lane  = threadIdx.x & 31;
    const int wave  = threadIdx.x >> 5;
    const int m0    = blockIdx.x * 16;
    const int e0    = wave * 128;
    const int row16 = lane & 15;
    const int half4 = (lane >> 4) * 4;   // 0 or 4: float4 base within the k-chunk

    const float* aRow = inp + (size_t)(m0 + row16) * D_ + half4;

    v8f acc[8] = {};

    for (int c = 0; c < D_; c += 8) {
        const v4f a4 = *(const v4f*)(aRow + c);
        const v2f a01 = a4.xy;
        const v2f a23 = a4.zw;
        #pragma unroll
        for (int j = 0; j < 8; ++j) {
            const v4f b4 = *(const v4f*)(wm + (size_t)(e0 + j * 16 + row16) * D_ + c + half4);
            acc[j] = __builtin_amdgcn_wmma_f32_16x16x4_f32(
                false, a01, false, b4.xy, (short)0, acc[j], false, false);
            acc[j] = __builtin_amdgcn_wmma_f32_16x16x4_f32(
                false, a23, false, b4.zw, (short)0, acc[j], false, false);
        }
    }

    const int crow = (lane >> 4) * 8;
    #pragma unroll
    for (int j = 0; j < 8; ++j) {
        const int col  = e0 + j * 16 + row16;
        const float bb = bs[col];
        #pragma unroll
        for (int v = 0; v < 8; ++v) {
            outS[(size_t)(m0 + crow + v) * D_ + col] = acc[j][v] + bb;
        }
    }
}

// ---------------------------------------------------------------------------
// Kernel 4: out[b,t,f,d] = sum_p Wt[f,p] * out_s[b,t+p,d] + bt[f]
//   One block per (b,t); 256 lanes x float4 span D=1024.
//   out_s (37.7 MB) is L2-resident (192 MB L2) -> 12x window re-reads hit L2.
//   Final stores are non-temporal: 434 MB that is never re-read; this write
//   stream is the HBM roofline floor (~19 us at 23.3 TB/s).
// ---------------------------------------------------------------------------
__global__ __launch_bounds__(256) void temporal_mix_kernel(
    const float* __restrict__ outS,  // [B, L, D]
    const float* __restrict__ Wt,    // [FUT, PAST]
    const float* __restrict__ bt,    // [FUT]
    float* __restrict__ out)         // [B, T, FUT, D]
{
    __shared__ float sWt[FUT_ * PAST_];
    __shared__ float sBt[FUT_];
    if (threadIdx.x < FUT_ * PAST_) sWt[threadIdx.x] = Wt[threadIdx.x];
    if (threadIdx.x < FUT_)         sBt[threadIdx.x] = bt[threadIdx.x];
    __syncthreads();

    const int btIdx = blockIdx.x;        // b*T + t
    const int b = btIdx / T_;
    const int t = btIdx - b * T_;
    const int d = threadIdx.x * 4;

    const float* base = outS + ((size_t)(b * L_ + t)) * D_ + d;
    v4f win[PAST_];
    #pragma unroll
    for (int p = 0; p < PAST_; ++p)
        win[p] = *(const v4f*)(base + (size_t)p * D_);

    float* obase = out + ((size_t)btIdx * FUT_) * D_ + d;
    #pragma unroll
    for (int f = 0; f < FUT_; ++f) {
        const float bb = sBt[f];
        v4f accv = { bb, bb, bb, bb };
        #pragma unroll
        for (int p = 0; p < PAST_; ++p) {
            accv += sWt[f * PAST_ + p] * win[p];
        }
        __builtin_nontemporal_store(accv, (v4f*)(obase + (size_t)f * D_));
    }
}

// ---------------------------------------------------------------------------
extern "C" void kernel_launch(void* const* d_in, const int* in_sizes, int n_in,
                              void* d_out, int out_size, void* d_ws, size_t ws_size,
                              hipStream_t stream) {
    const float* inp = (const float*)d_in[0];
    const float* Ws  = (const float*)d_in[1];
    const float* bs  = (const float*)d_in[2];
    const float* adj = (const float*)d_in[3];
    const float* Wt  = (const float*)d_in[4];
    const float* btp = (const float*)d_in[5];

    float* out   = (float*)d_out;
    float* wtOut = out + (size_t)B_ * T_ * FUT_ * D_;   // tuple output #2
    float* wmOut = wtOut + FUT_ * PAST_;                // tuple output #3
    float* outS  = (float*)d_ws;                        // 9216*1024 fp32 = 37.7 MB

    wm_mask_kernel<<<(D_ * D_) / 256, 256, 0, stream>>>(Ws, adj, wmOut);
    wt_copy_kernel<<<1, 256, 0, stream>>>(Wt, wtOut);
    spatial_gemm_kernel<<<(B_ * L_) / 16, 256, 0, stream>>>(inp, wmOut, bs, outS);
    temporal_mix_kernel<<<B_ * T_, 256, 0, stream>>>(outS, Wt, btp, out);
}